// MyRnn_89481348645115
// MI455X (gfx1250) — compile-verified
//
#include <hip/hip_runtime.h>
#include <math.h>

typedef __attribute__((ext_vector_type(2))) float v2f;
typedef __attribute__((ext_vector_type(8))) float v8f;

#define Bc 64
#define Tc 512
#define Ic 1024
#define Hc 1024
#define BTc (Bc * Tc)

// D = A(16x4 f32) * B(4x16 f32) + C(16x16 f32), full fp32 matrix pipe.
#define WMMA_F32(a, b, c) \
  __builtin_amdgcn_wmma_f32_16x16x4_f32(false, (a), false, (b), (short)0, (c), false, false)

// ---------------------------------------------------------------------------
// Phase 1: xw[bt][h] = sum_i x[bt][i] * W[h][i] + bW[h]   (written into out1)
// WG = 128 threads = 4 waves in a 2x2 arrangement; each wave computes a
// 64(M) x 64(N) register tile (16 accumulators). Per K=4 step: 4 A-frags +
// 4 B-frags (8 x b64 loads) feed 16 WMMAs -> 4 B/lane/WMMA operand traffic.
// grid = (H/128, BT/128)
// ---------------------------------------------------------------------------
__global__ __launch_bounds__(128) void xw_gemm_kernel(
    const float* __restrict__ x, const float* __restrict__ W,
    const float* __restrict__ bW, float* __restrict__ out1) {
  const int lane = threadIdx.x & 31;
  const int wave = threadIdx.x >> 5;
  const int r    = lane & 15;
  const int hi   = lane >> 4;
  const int koff = hi * 2;   // lanes 16-31 hold K+2,K+3 (ISA 16x4 A / 4x16 B layout)
  const int mrow = hi * 8;   // C layout: lanes 16-31 hold M = i + 8

  const int m0 = blockIdx.y * 128 + (wave & 1) * 64;
  const int n0 = blockIdx.x * 128 + (wave >> 1) * 64;

  const float* aP[4];
  const float* bP[4];
#pragma unroll
  for (int i = 0; i < 4; ++i) {
    aP[i] = x + (size_t)(m0 + i * 16 + r) * Ic + koff;  // A[m][k..k+1]
    bP[i] = W + (size_t)(n0 + i * 16 + r) * Ic + koff;  // B[k][n] = W[n][k]
  }

  v8f acc[4][4];
#pragma unroll
  for (int i = 0; i < 4; ++i)
#pragma unroll
    for (int j = 0; j < 4; ++j) acc[i][j] = (v8f){};

#pragma unroll 2
  for (int k = 0; k < Ic; k += 4) {
    v2f a[4], b[4];
#pragma unroll
    for (int i = 0; i < 4; ++i) a[i] = *(const v2f*)(aP[i] + k);
#pragma unroll
    for (int j = 0; j < 4; ++j) b[j] = *(const v2f*)(bP[j] + k);
#pragma unroll
    for (int i = 0; i < 4; ++i)
#pragma unroll
      for (int j = 0; j < 4; ++j) acc[i][j] = WMMA_F32(a[i], b[j], acc[i][j]);
  }

#pragma unroll
  for (int j = 0; j < 4; ++j) {
    const int n = n0 + j * 16 + r;
    const float bias = bW[n];
#pragma unroll
    for (int i = 0; i < 4; ++i) {
      float* o = out1 + (size_t)(m0 + i * 16 + mrow) * Hc + n;
#pragma unroll
      for (int e = 0; e < 8; ++e) {
        o[(size_t)e * Hc] = acc[i][j][e] + bias;
      }
    }
  }
}

// ---------------------------------------------------------------------------
// Phase 2: persistent scan kernel.
//   for t: h3 = tanh(xw_t + state); out1[:,t,:] = h3;  state = h3 @ U.T + bU
// grid = 64 WGs (4 M-tiles x 16 N-strips), 128 threads = 4 waves.
// Each WG owns one 16x64 state tile (kept in LDS); waves K-split the 1024-deep
// reduction 4 ways, partials reduced through LDS. One device-wide atomic
// barrier per step (all 64 WGs resident; U stays hot in the 192MB L2).
// ---------------------------------------------------------------------------
__global__ __launch_bounds__(128) void rnn_scan_kernel(
    const float* __restrict__ U, const float* __restrict__ bU,
    float* __restrict__ out1, float* __restrict__ out2,
    unsigned* __restrict__ bar) {
  __shared__ float s_state[16 * 64];
  __shared__ float s_part[4][16 * 64];

  const int tid  = threadIdx.x;
  const int lane = tid & 31;
  const int wave = tid >> 5;
  const int r    = lane & 15;
  const int hi   = lane >> 4;
  const int koff = hi * 2;
  const int mrow = hi * 8;

  const int m0 = (blockIdx.x & 3) * 16;   // M tile
  const int n0 = (blockIdx.x >> 2) * 64;  // N strip
  const int kb = wave * 256;              // K quarter for this wave

  // Per-thread 8-element slice of the 16x64 tile (one row fragment).
  const int e0   = tid * 8;
  const int mloc = e0 >> 6;
  const int nloc = e0 & 63;

  float bu[8];
#pragma unroll
  for (int q = 0; q < 8; ++q) bu[q] = bU[n0 + nloc + q];

  for (int e = tid; e < 1024; e += 128) s_state[e] = 0.0f;  // t0 = zeros
  __syncthreads();

  const unsigned NWG = gridDim.x;

  for (int t = 0; t < Tc; ++t) {
    // ---- h3 = tanh(xw_t + state), written in-place over xw in out1 ----
    size_t xwIdx = ((size_t)(m0 + mloc) * Tc + t) * Hc + (n0 + nloc);
    float h[8];
#pragma unroll
    for (int q = 0; q < 8; ++q) {
      h[q] = tanhf(out1[xwIdx + q] + s_state[e0 + q]);
      out1[xwIdx + q] = h[q];
    }
    if (t == Tc - 1) {
      size_t o2 = (size_t)(m0 + mloc) * Hc + (n0 + nloc);
#pragma unroll
      for (int q = 0; q < 8; ++q) out2[o2 + q] = h[q];
      break;  // last t_next is unused by the reference
    }

    // ---- device-wide barrier: all of h3(t) visible before the GEMM ----
    __threadfence();
    __syncthreads();
    if (tid == 0) {
      unsigned gen = __hip_atomic_load(&bar[1], __ATOMIC_ACQUIRE, __HIP_MEMORY_SCOPE_AGENT);
      unsigned arr = __hip_atomic_fetch_add(&bar[0], 1u, __ATOMIC_ACQ_REL, __HIP_MEMORY_SCOPE_AGENT);
      if (arr == NWG - 1u) {
        __hip_atomic_store(&bar[0], 0u, __ATOMIC_RELAXED, __HIP_MEMORY_SCOPE_AGENT);
        __hip_atomic_fetch_add(&bar[1], 1u, __ATOMIC_RELEASE, __HIP_MEMORY_SCOPE_AGENT);
      } else {
        while (__hip_atomic_load(&bar[1], __ATOMIC_ACQUIRE, __HIP_MEMORY_SCOPE_AGENT) == gen) {
          __builtin_amdgcn_s_sleep(2);
        }
      }
    }
    __syncthreads();

    // ---- partial GEMM over this wave's K quarter: h3 @ U.T ----
    const float* aB  = out1 + ((size_t)(m0 + r) * Tc + t) * Hc + kb + koff;
    const float* bB0 = U + (size_t)(n0 +  0 + r) * Hc + kb + koff;  // B[k][n] = U[n][k]
    const float* bB1 = U + (size_t)(n0 + 16 + r) * Hc + kb + koff;
    const float* bB2 = U + (size_t)(n0 + 32 + r) * Hc + kb + koff;
    const float* bB3 = U + (size_t)(n0 + 48 + r) * Hc + kb + koff;

    v8f c0 = {}, c1 = {}, c2 = {}, c3 = {};
#pragma unroll 4
    for (int k = 0; k < 256; k += 4) {
      v2f a  = *(const v2f*)(aB + k);
      v2f b0 = *(const v2f*)(bB0 + k);
      v2f b1 = *(const v2f*)(bB1 + k);
      v2f b2 = *(const v2f*)(bB2 + k);
      v2f b3 = *(const v2f*)(bB3 + k);
      c0 = WMMA_F32(a, b0, c0);
      c1 = WMMA_F32(a, b1, c1);
      c2 = WMMA_F32(a, b2, c2);
      c3 = WMMA_F32(a, b3, c3);
    }

    v8f cc[4] = {c0, c1, c2, c3};
#pragma unroll
    for (int j = 0; j < 4; ++j)
#pragma unroll
      for (int i = 0; i < 8; ++i)
        s_part[wave][(mrow + i) * 64 + j * 16 + r] = cc[j][i];
    __syncthreads();

    // ---- reduce 4 K-partials + bias into the new state ----
#pragma unroll
    for (int q = 0; q < 8; ++q) {
      int e = e0 + q;
      s_state[e] = bu[q] + s_part[0][e] + s_part[1][e] + s_part[2][e] + s_part[3][e];
    }
    __syncthreads();
  }
}

__global__ void init_bar_kernel(unsigned* bar) {
  if (threadIdx.x < 2) bar[threadIdx.x] = 0u;
}

extern "C" void kernel_launch(void* const* d_in, const int* in_sizes, int n_in,
                              void* d_out, int out_size, void* d_ws, size_t ws_size,
                              hipStream_t stream) {
  const float* x  = (const float*)d_in[0];
  const float* W  = (const float*)d_in[1];
  const float* bW = (const float*)d_in[2];
  const float* U  = (const float*)d_in[3];
  const float* bU = (const float*)d_in[4];

  float* out1 = (float*)d_out;                       // (B, T, H)
  float* out2 = out1 + (size_t)BTc * Hc;             // (B, 1, H)
  unsigned* bar = (unsigned*)d_ws;                   // [count, generation]

  hipLaunchKernelGGL(init_bar_kernel, dim3(1), dim3(32), 0, stream, bar);
  hipLaunchKernelGGL(xw_gemm_kernel, dim3(Hc / 128, BTc / 128), dim3(128), 0, stream,
                     x, W, bW, out1);
  hipLaunchKernelGGL(rnn_scan_kernel, dim3(64), dim3(128), 0, stream,
                     U, bU, out1, out2, bar);
}